// HybridImageFittingModel_65077344469002
// MI455X (gfx1250) — compile-verified
//
#include <hip/hip_runtime.h>
#include <hip/hip_bf16.h>
#include <math.h>

typedef __attribute__((ext_vector_type(2))) float v2f;
typedef __attribute__((ext_vector_type(8))) float v8f;

#define NQ 6
#define DIM 64
#define B_TOTAL 16384
#define BN_EPS 1e-5f

// workspace float offsets
#define G_OFF    0
#define WR_OFF   64
#define WI_OFF   (64 + 4096)
#define SUM_OFF  (64 + 8192)        // 12 floats used
#define PART_OFF (SUM_OFF + 16)     // 64*12 floats

__device__ __forceinline__ float2 cmul(float2 a, float2 b) {
    return make_float2(a.x * b.x - a.y * b.y, a.x * b.y + a.y * b.x);
}
__device__ __forceinline__ float2 cadd(float2 a, float2 b) {
    return make_float2(a.x + b.x, a.y + b.y);
}

// index with bit `pos` cleared, other 5 bits from p (p in [0,32))
__device__ __forceinline__ int pair_lo(int p, int pos) {
    int low = p & ((1 << pos) - 1);
    return ((p >> pos) << (pos + 1)) | low;
}

// 6-bit index with fixed bits bitA@posA, bitB@posB, other 4 bits from p
__device__ __forceinline__ int spread2(int p, int posA, int posB, int bitA, int bitB) {
    int k = 0, b = 0;
    #pragma unroll
    for (int pos = 0; pos < 6; ++pos) {
        int bit;
        if (pos == posA)      bit = bitA;
        else if (pos == posB) bit = bitB;
        else { bit = (p >> b) & 1; ++b; }
        k |= bit << pos;
    }
    return k;
}

struct CU { float2 u00, u01, u10, u11; };

// PennyLane Rot = RZ(omega) RY(theta) RZ(phi)
__device__ __forceinline__ CU make_rot(float ph, float th, float om) {
    float s, c;
    __sincosf(0.5f * th, &s, &c);
    float a1s, a1c, a2s, a2c;
    __sincosf(-0.5f * (ph + om), &a1s, &a1c);   // ep = (a1c, a1s)
    __sincosf(-0.5f * (ph - om), &a2s, &a2c);   // em = (a2c, a2s)
    CU u;
    u.u00 = make_float2(a1c * c,  a1s * c);     // ep*c
    u.u01 = make_float2(-a2c * s, a2s * s);     // -conj(em)*s
    u.u10 = make_float2(a2c * s,  a2s * s);     // em*s
    u.u11 = make_float2(a1c * c, -a1s * c);     // conj(ep)*c
    return u;
}

// ---------------------------------------------------------------------------
// Prep (1 block, 64 threads): psi1 = U1|0>, build W = U2*diag(psi1), g vector
// ---------------------------------------------------------------------------
__global__ __launch_bounds__(64)
void prep_kernel(const float* __restrict__ w1q,   // [3][6][3]
                 const float* __restrict__ w2q,   // [1][6][3]
                 const float* __restrict__ wlin2, // [1][6]
                 float* __restrict__ ws) {
    __shared__ float2 S[DIM * DIM];   // 32 KB: U2 being built, columns = basis inputs
    __shared__ float2 psi[DIM];
    const int tid = threadIdx.x;

    // ---- psi1: apply 9 SEL(weights1) layers to |0...0>
    if (tid < DIM) psi[tid] = make_float2(tid == 0 ? 1.f : 0.f, 0.f);
    __syncthreads();
    for (int rep = 0; rep < 3; ++rep) {
        for (int l = 0; l < 3; ++l) {
            const int r = l + 1;  // l % 5 + 1 for l in {0,1,2}
            for (int j = 0; j < NQ; ++j) {
                const float* w = &w1q[(l * NQ + j) * 3];
                CU u = make_rot(w[0], w[1], w[2]);
                if (tid < 32) {
                    int pos = 5 - j;
                    int k0 = pair_lo(tid, pos), k1 = k0 | (1 << pos);
                    float2 a0 = psi[k0], a1 = psi[k1];
                    psi[k0] = cadd(cmul(u.u00, a0), cmul(u.u01, a1));
                    psi[k1] = cadd(cmul(u.u10, a0), cmul(u.u11, a1));
                }
                __syncthreads();
            }
            for (int j = 0; j < NQ; ++j) {            // CNOT ring, range r
                int pc = 5 - j, pt = 5 - ((j + r) % NQ);
                if (tid < 16) {
                    int k  = spread2(tid, pc, pt, 1, 0);
                    int k2 = k | (1 << pt);
                    float2 t = psi[k]; psi[k] = psi[k2]; psi[k2] = t;
                }
                __syncthreads();
            }
        }
    }

    // ---- U2: SEL(weights2) single layer (Rot each wire, then CZ ring r=1)
    for (int q = tid; q < DIM * DIM; q += 64)
        S[q] = make_float2((q / DIM) == (q % DIM) ? 1.f : 0.f, 0.f);
    __syncthreads();
    for (int j = 0; j < NQ; ++j) {
        const float* w = &w2q[j * 3];
        CU u = make_rot(w[0], w[1], w[2]);
        int pos = 5 - j;
        for (int q = tid; q < 32 * DIM; q += 64) {
            int p = q / DIM, col = q % DIM;
            int k0 = pair_lo(p, pos), k1 = k0 | (1 << pos);
            float2 a0 = S[k0 * DIM + col], a1 = S[k1 * DIM + col];
            S[k0 * DIM + col] = cadd(cmul(u.u00, a0), cmul(u.u01, a1));
            S[k1 * DIM + col] = cadd(cmul(u.u10, a0), cmul(u.u11, a1));
        }
        __syncthreads();
    }
    for (int j = 0; j < NQ; ++j) {                    // CZ ring range 1
        int pc = 5 - j, pt = 5 - ((j + 1) % NQ);
        for (int q = tid; q < 16 * DIM; q += 64) {
            int p = q / DIM, col = q % DIM;
            int k = spread2(p, pc, pt, 1, 1);
            S[k * DIM + col].x = -S[k * DIM + col].x;
            S[k * DIM + col].y = -S[k * DIM + col].y;
        }
        __syncthreads();
    }

    // ---- W[k][l] = U2[k][l] * psi1[l]  (row-major, split re/im)
    for (int q = tid; q < DIM * DIM; q += 64) {
        int l = q % DIM;
        float2 w = cmul(S[q], psi[l]);
        ws[WR_OFF + q] = w.x;
        ws[WI_OFF + q] = w.y;
    }

    // ---- g[k] = sum_j w_lin2[j] * (1 - 2*bit_{5-j}(k))
    if (tid < DIM) {
        float g = 0.f;
        #pragma unroll
        for (int j = 0; j < NQ; ++j) {
            float sgn = ((tid >> (5 - j)) & 1) ? -1.f : 1.f;
            g += wlin2[j] * sgn;
        }
        ws[G_OFF + tid] = g;
    }
}

// ---------------------------------------------------------------------------
// BatchNorm statistics: deterministic two-pass reduction (no atomics)
// ---------------------------------------------------------------------------
__global__ __launch_bounds__(256)
void bn_partial_kernel(const float* __restrict__ x, const float* __restrict__ w1,
                       const float* __restrict__ b1, float* __restrict__ ws) {
    __shared__ float red[256];
    const int s = blockIdx.x * 256 + threadIdx.x;
    const float x0 = x[2 * s], x1 = x[2 * s + 1];
    float vals[12];
    #pragma unroll
    for (int j = 0; j < NQ; ++j) {
        float h = x0 * w1[2 * j] + x1 * w1[2 * j + 1] + b1[j];
        vals[j] = h;
        vals[6 + j] = h * h;
    }
    for (int c = 0; c < 12; ++c) {
        red[threadIdx.x] = vals[c];
        __syncthreads();
        for (int off = 128; off > 0; off >>= 1) {
            if (threadIdx.x < off) red[threadIdx.x] += red[threadIdx.x + off];
            __syncthreads();
        }
        if (threadIdx.x == 0) ws[PART_OFF + blockIdx.x * 12 + c] = red[0];
        __syncthreads();
    }
}

__global__ __launch_bounds__(32)
void bn_final_kernel(float* __restrict__ ws) {
    const int c = threadIdx.x;
    if (c < 12) {
        float s = 0.f;
        for (int i = 0; i < 64; ++i) s += ws[PART_OFF + i * 12 + c];
        ws[SUM_OFF + c] = s;
    }
}

// ---------------------------------------------------------------------------
// Main: per-wave 16-sample tile; complex 64x64 matvec via V_WMMA_F32_16X16X4_F32
// ---------------------------------------------------------------------------
__global__ __launch_bounds__(128)
void qcircuit_main_kernel(const float* __restrict__ x,  const float* __restrict__ w1,
                          const float* __restrict__ b1, const float* __restrict__ gamma,
                          const float* __restrict__ beta, const float* __restrict__ blin2,
                          const float* __restrict__ ws,  float* __restrict__ out) {
    __shared__ float hbuf[4][16][NQ];
    __shared__ float Dr[4][16 * DIM];   // cos(phi)
    __shared__ float Di[4][16 * DIM];   // sin(phi)
    __shared__ float Dn[4][16 * DIM];   // -sin(phi)  (f32 WMMA has no A-negate)
    __shared__ float red[4][16][16];
    __shared__ float gsh[DIM];

    const int tid  = threadIdx.x;
    const int wv   = tid >> 5;
    const int ln   = tid & 31;
    const int base = (blockIdx.x * 4 + wv) * 16;

    if (tid < DIM) gsh[tid] = ws[G_OFF + tid];

    // ---- BN'd angles h for this wave's 16 samples
    for (int idx = ln; idx < 16 * NQ; idx += 32) {
        int sl = idx / NQ, j = idx % NQ;
        int s  = base + sl;
        float m   = ws[SUM_OFF + j] * (1.f / B_TOTAL);
        float var = ws[SUM_OFF + 6 + j] * (1.f / B_TOTAL) - m * m;
        float inv = rsqrtf(var + BN_EPS);
        float h   = x[2 * s] * w1[2 * j] + x[2 * s + 1] * w1[2 * j + 1] + b1[j];
        hbuf[wv][sl][j] = (h - m) * inv * gamma[j] + beta[j];
    }
    __syncthreads();

    // ---- phase vectors d = exp(i*phi), row-major [16 samples][64]
    for (int it = 0; it < 32; ++it) {
        int p  = it * 32 + ln;
        int sl = p >> 6, l = p & 63;
        float phi = 0.f;
        #pragma unroll
        for (int j = 0; j < NQ; ++j) {
            float sgn = ((l >> (5 - j)) & 1) ? 1.f : -1.f;  // RZ: bit1 -> +a/2
            phi += sgn * hbuf[wv][sl][j];
        }
        phi *= 0.5f;
        float sp, cp;
        __sincosf(phi, &sp, &cp);
        Dr[wv][p] = cp; Di[wv][p] = sp; Dn[wv][p] = -sp;
    }
    __syncthreads();

    // ---- V[s][k] = sum_l D[s][l] * W[k][l]  (complex), via f32 WMMA K=4
    const int hi = ln >> 4;      // lane group selects K pair {0,1} vs {2,3}
    const int lo = ln & 15;      // A: sample row / B,C: column
    float part[8];
    #pragma unroll
    for (int r = 0; r < 8; ++r) part[r] = 0.f;

    const float* WrG = ws + WR_OFF;
    const float* WiG = ws + WI_OFF;

    for (int ct = 0; ct < 4; ++ct) {
        v8f accR = {}; v8f accI = {};
        const int wrow = ct * 16 + lo;                 // output amplitude index
        #pragma unroll
        for (int kt = 0; kt < 16; ++kt) {
            const int kk = kt * 4 + 2 * hi;
            // A fragments (LDS, 8B-aligned pairs)
            const float* dr = &Dr[wv][lo * DIM + kk];
            const float* di = &Di[wv][lo * DIM + kk];
            const float* dn = &Dn[wv][lo * DIM + kk];
            v2f aR; aR[0] = dr[0]; aR[1] = dr[1];
            v2f aI; aI[0] = di[0]; aI[1] = di[1];
            v2f aN; aN[0] = dn[0]; aN[1] = dn[1];
            // B fragments: B[l][k] = W[k][l] -> row `wrow` of W (L2-resident)
            const float* br = &WrG[wrow * DIM + kk];
            const float* bi = &WiG[wrow * DIM + kk];
            v2f bR; bR[0] = br[0]; bR[1] = br[1];
            v2f bI; bI[0] = bi[0]; bI[1] = bi[1];

            // Vr += Dr*Wr - Di*Wi ; Vi += Dr*Wi + Di*Wr
            accR = __builtin_amdgcn_wmma_f32_16x16x4_f32(false, aR, false, bR, (short)0, accR, false, false);
            accR = __builtin_amdgcn_wmma_f32_16x16x4_f32(false, aN, false, bI, (short)0, accR, false, false);
            accI = __builtin_amdgcn_wmma_f32_16x16x4_f32(false, aR, false, bI, (short)0, accI, false, false);
            accI = __builtin_amdgcn_wmma_f32_16x16x4_f32(false, aI, false, bR, (short)0, accI, false, false);
        }
        // C layout: acc[r] holds (row = r + 8*hi, col = lo); weight by g[col]
        const float gc = gsh[ct * 16 + lo];
        #pragma unroll
        for (int r = 0; r < 8; ++r)
            part[r] += gc * (accR[r] * accR[r] + accI[r] * accI[r]);
    }

    // ---- reduce |v|^2 contributions per sample across the 16 column-lanes
    #pragma unroll
    for (int r = 0; r < 8; ++r)
        red[wv][r + 8 * hi][lo] = part[r];
    __syncthreads();

    if (ln < 16) {
        float s = 0.f;
        #pragma unroll
        for (int c = 0; c < 16; ++c) s += red[wv][ln][c];
        out[base + ln] = s + blin2[0];
    }
}

// ---------------------------------------------------------------------------
extern "C" void kernel_launch(void* const* d_in, const int* in_sizes, int n_in,
                              void* d_out, int out_size, void* d_ws, size_t ws_size,
                              hipStream_t stream) {
    const float* x        = (const float*)d_in[0];
    const float* w_lin1   = (const float*)d_in[1];
    const float* b_lin1   = (const float*)d_in[2];
    const float* bn_gamma = (const float*)d_in[3];
    const float* bn_beta  = (const float*)d_in[4];
    const float* weights1 = (const float*)d_in[5];
    const float* weights2 = (const float*)d_in[6];
    const float* w_lin2   = (const float*)d_in[7];
    const float* b_lin2   = (const float*)d_in[8];
    float* out = (float*)d_out;
    float* ws  = (float*)d_ws;

    prep_kernel<<<1, 64, 0, stream>>>(weights1, weights2, w_lin2, ws);
    bn_partial_kernel<<<B_TOTAL / 256, 256, 0, stream>>>(x, w_lin1, b_lin1, ws);
    bn_final_kernel<<<1, 32, 0, stream>>>(ws);
    qcircuit_main_kernel<<<B_TOTAL / 64, 128, 0, stream>>>(
        x, w_lin1, b_lin1, bn_gamma, bn_beta, b_lin2, ws, out);
}